// GAE_43602507989602
// MI455X (gfx1250) — compile-verified
//
#include <hip/hip_runtime.h>

// ---------------------------------------------------------------------------
// Problem constants (from the reference)
// ---------------------------------------------------------------------------
#define NN      128               // N
#define CC      256               // GC_CH
#define KTOT    (NN * CC)         // 32768 = flat inner dim of final GEMM
#define DH      512               // DEC_H
#define BATCH   16                // B
#define NKSLICE 16                // K-slices for parallelism
#define KSLICE  (KTOT / NKSLICE)  // 2048 K per block
#define KWAVE   (KSLICE / 8)      // 256 K per wave (8 waves / block)
#define BN_EPS  1e-3f

typedef float v2f __attribute__((ext_vector_type(2)));
typedef float v8f __attribute__((ext_vector_type(8)));

// ---------------------------------------------------------------------------
// Kernel 1: v[i*256+c] = PReLU( BN(gc_b)[c], prelu_alpha[i,c] )
// (encoder / deconv / softmax are provably dead: adj = sum(softmax) == 1 > 0
//  for any finite input, so the max-aggregation mask is all-true; node
//  features are literal zeros, so hh depends only on the channel index)
// ---------------------------------------------------------------------------
__global__ __launch_bounds__(256) void gae_prep_v(
    const float* __restrict__ gc_b,  const float* __restrict__ bn_gamma,
    const float* __restrict__ bn_beta, const float* __restrict__ bn_mean,
    const float* __restrict__ bn_var,  const float* __restrict__ prelu_alpha,
    float* __restrict__ v)
{
    const int idx = blockIdx.x * 256 + threadIdx.x;   // = i*CC + c
    const int c   = idx & (CC - 1);
    const float f = bn_gamma[c] * (gc_b[c] - bn_mean[c]) *
                    rsqrtf(bn_var[c] + BN_EPS) + bn_beta[c];
    const float a = prelu_alpha[idx];                 // (N, C) row-major == flat idx
    v[idx] = (f > 0.0f) ? f : a * f;
}

// ---------------------------------------------------------------------------
// Kernel 2: partial GEMM  flat[16 x 32768] @ W_dec[32768 x 512]
// All 16 rows of flat equal v, so the A 16x4 fragment is a broadcast of v.
// grid = (DH/16 column tiles, NKSLICE k-slices), block = 256 (8 wave32)
// -> 512 blocks / 4096 waves in flight to cover HBM latency.
// Uses V_WMMA_F32_16X16X4_F32 (exact fp32 path, M=16 == batch).
// B loads use a single stepped base pointer so all offsets fold into the
// signed-24-bit instruction IOFFSET (no per-load 64-bit address VALU).
// ---------------------------------------------------------------------------
__global__ __launch_bounds__(256) void gae_gemm_partial(
    const float* __restrict__ v, const float* __restrict__ W,
    float* __restrict__ partials)
{
    __shared__ float red[8][16];

    const int wave = threadIdx.x >> 5;
    const int lane = threadIdx.x & 31;
    const int col  = (blockIdx.x << 4) + (lane & 15);
    // A 16x4 f32 layout: lanes 0-15 hold K=0,1 ; lanes 16-31 hold K=2,3.
    const int koff = (lane < 16) ? 0 : 2;

    const int k0 = blockIdx.y * KSLICE + wave * KWAVE + koff;

    const float* __restrict__ vp = v + k0;                      // A source
    const float* __restrict__ Wp = W + (size_t)k0 * DH + col;   // B source

    v8f acc = {};
    #pragma unroll 1
    for (int kk = 0; kk < KWAVE; kk += 16) {                    // 4 WMMA / iter
        #pragma unroll
        for (int j = 0; j < 4; ++j) {
            // A fragment: rows identical -> broadcast v[k..k+1] (8B aligned).
            v2f a = *(const v2f*)(vp + j * 4);
            // B fragment: same K-striping; immediate-offset column loads.
            v2f b;
            b.x = Wp[j * 4 * DH];           // byte offsets 0 / 8192 / 16384 / 24576
            b.y = Wp[j * 4 * DH + DH];      // +2048B
            acc = __builtin_amdgcn_wmma_f32_16x16x4_f32(
                      /*neg_a=*/false, a, /*neg_b=*/false, b,
                      /*c_mod=*/(short)0, acc,
                      /*reuse_a=*/false, /*reuse_b=*/false);
        }
        vp += 16;
        Wp += 16 * DH;                      // one 64-bit pointer bump per group
    }

    // C layout: VGPR0, lanes 0-15 == (M=0, N=lane). All rows identical; keep M=0.
    if (lane < 16) red[wave][lane] = acc[0];
    __syncthreads();

    if (threadIdx.x < 16) {
        float s = 0.0f;
        #pragma unroll
        for (int w = 0; w < 8; ++w) s += red[w][threadIdx.x];
        partials[blockIdx.y * DH + (blockIdx.x << 4) + threadIdx.x] = s;
    }
}

// ---------------------------------------------------------------------------
// Kernel 3: reduce K-slices, add bias, ReLU, broadcast to the 16 batch rows.
// ---------------------------------------------------------------------------
__global__ __launch_bounds__(256) void gae_finalize(
    const float* __restrict__ partials, const float* __restrict__ b_dec,
    float* __restrict__ out)
{
    const int d = blockIdx.x * 256 + threadIdx.x;
    float s = b_dec[d];
    #pragma unroll
    for (int sl = 0; sl < NKSLICE; ++sl) s += partials[sl * DH + d];
    const float y = fmaxf(s, 0.0f);
    #pragma unroll
    for (int b = 0; b < BATCH; ++b) out[b * DH + d] = y;
}

// ---------------------------------------------------------------------------
// Launch
// ---------------------------------------------------------------------------
extern "C" void kernel_launch(void* const* d_in, const int* in_sizes, int n_in,
                              void* d_out, int out_size, void* d_ws, size_t ws_size,
                              hipStream_t stream)
{
    (void)in_sizes; (void)n_in; (void)out_size; (void)ws_size;

    // setup_inputs order:
    // 0 embedding, 1 W_enc, 2 b_enc, 3 K_deconv, 4 b_deconv, 5 gc_W,
    // 6 gc_b, 7 bn_gamma, 8 bn_beta, 9 bn_mean, 10 bn_var,
    // 11 prelu_alpha, 12 W_dec, 13 b_dec
    const float* gc_b     = (const float*)d_in[6];
    const float* bn_gamma = (const float*)d_in[7];
    const float* bn_beta  = (const float*)d_in[8];
    const float* bn_mean  = (const float*)d_in[9];
    const float* bn_var   = (const float*)d_in[10];
    const float* alpha    = (const float*)d_in[11];
    const float* W_dec    = (const float*)d_in[12];
    const float* b_dec    = (const float*)d_in[13];
    float*       out      = (float*)d_out;

    // Workspace layout: v[32768] floats, then partials[NKSLICE*DH] floats.
    float* v        = (float*)d_ws;
    float* partials = v + KTOT;

    gae_prep_v<<<dim3(KTOT / 256), dim3(256), 0, stream>>>(
        gc_b, bn_gamma, bn_beta, bn_mean, bn_var, alpha, v);

    gae_gemm_partial<<<dim3(DH / 16, NKSLICE), dim3(256), 0, stream>>>(
        v, W_dec, partials);

    gae_finalize<<<dim3(DH / 256), dim3(256), 0, stream>>>(
        partials, b_dec, out);
}